// OhemCrossEntropy_68753836474471
// MI455X (gfx1250) — compile-verified
//
#include <hip/hip_runtime.h>

// ---------------- problem constants (from reference) ----------------
#define IGNORE_LABEL 255
#define THRESH_F     0.7f
#define MIN_KEPT     100000
#define NCLS         19
#define NBATCH       8
#define HWSZ         (512 * 1024)          // 524288
#define NPIX         (NBATCH * HWSZ)       // 4194304

// ---------------- tuning ----------------
#define NB      2048                       // prob-histogram bins (res 4.9e-4)
#define BLK     256
#define WPB     (BLK / 32)                 // 8 waves per block (wave32)
#define GRID1   1024
#define NWAVES  (GRID1 * WPB)              // 8192
#define STEPS   (NPIX / (NWAVES * 32))     // 16 tiles of 32 px per wave
#define GRID3   1024

// ---------------- ws layout (32-bit word offsets) ----------------
#define WS_THR    0                        // float threshold
#define WS_NVALID 1                        // uint n_valid
#define WS_HIST   8                        // NB u32
#define WS_PSUM   2304                     // GRID3 floats
#define WS_PCNT   3328                     // GRID3 u32
#define WS_LOGP   4608                     // NPIX floats (16B aligned)

// ---------------- CDNA5 async global->LDS path ----------------
#if defined(__gfx1250__) &&                                        \
    __has_builtin(__builtin_amdgcn_global_load_async_to_lds_b128) && \
    __has_builtin(__builtin_amdgcn_s_wait_asynccnt)
#define USE_ASYNC 1
#else
#define USE_ASYNC 0
#endif

typedef int v4i __attribute__((vector_size(16)));
typedef __attribute__((address_space(1))) v4i* g_v4i_ptr;  // global int4*
typedef __attribute__((address_space(3))) v4i* l_v4i_ptr;  // LDS int4*

__device__ __attribute__((always_inline)) inline
void async_cp16(const float* g, float* l) {
#if USE_ASYNC
  __builtin_amdgcn_global_load_async_to_lds_b128(
      (g_v4i_ptr)(g), (l_v4i_ptr)(l), 0, 0);
#else
  *(float4*)(l) = *(const float4*)(g);
#endif
}

template <int N>
__device__ __attribute__((always_inline)) inline void async_wait() {
#if USE_ASYNC
  __builtin_amdgcn_s_wait_asynccnt(N);
#endif
  __asm__ volatile("" ::: "memory");  // keep LDS reads below the wait
}

// ---------------- kernel 1: per-pixel CE + prob histogram ----------------
// Each wave owns a 19x32 fp32 LDS slice, double buffered, filled by async
// b128 copies; no block barriers in the hot loop (per-wave data only).
__global__ __launch_bounds__(BLK) void k_ce(const float* __restrict__ score,
                                            const int* __restrict__ target,
                                            float* __restrict__ logp_out,
                                            unsigned* __restrict__ ghist) {
  __shared__ float buf[2][WPB][NCLS * 32];   // 38912 B
  __shared__ unsigned hist[NB];              // 8192 B
  const int tid = threadIdx.x;
  const int lane = tid & 31;
  const int wv = tid >> 5;

  // fully unrolled zeroing: 8 unconditional constant-offset LDS stores
#pragma unroll
  for (int i = 0; i < NB / BLK; ++i) hist[tid + i * BLK] = 0u;
  __syncthreads();

  const int gw = blockIdx.x * WPB + wv;      // global wave id

  auto issue = [&](int s) {
    const int p0 = (gw + s * NWAVES) * 32;   // tile base pixel (mult of 32)
    const int b  = p0 / HWSZ;                // HWSZ % 32 == 0 -> tile in one image
    const int hw = p0 - b * HWSZ;
    const float* gbase = score + (size_t)b * NCLS * HWSZ + hw;
    float* lbase = &buf[s & 1][wv][0];
    // 19 rows * 8 float4 = 152 chunks. 5 issue slots per lane; slot 4 is
    // EXEC-predicated (lanes 0..23) -> still exactly 5 async instructions
    // per wave (uniform), with zero duplicate traffic.
#pragma unroll
    for (int i = 0; i < 5; ++i) {
      const int idx = lane + i * 32;
      if (idx < NCLS * 8) {
        const int c = idx >> 3, q = idx & 7;
        async_cp16(gbase + (size_t)c * HWSZ + q * 4, lbase + c * 32 + q * 4);
      }
    }
  };

  issue(0);
  for (int s = 0; s < STEPS; ++s) {
    if (s + 1 < STEPS) { issue(s + 1); async_wait<5>(); }
    else               { async_wait<0>(); }

    const float* L = &buf[s & 1][wv][0];
    const int p = (gw + s * NWAVES) * 32 + lane;
    const int t = target[p];
    const bool valid = (t != IGNORE_LABEL);

    // one LDS pass into registers; reuse for max and exp-sum
    float v[NCLS];
#pragma unroll
    for (int c = 0; c < NCLS; ++c) v[c] = L[c * 32 + lane];
    float m = v[0];
#pragma unroll
    for (int c = 1; c < NCLS; ++c) m = fmaxf(m, v[c]);
    const float xt = L[(valid ? t : 0) * 32 + lane];  // dynamic-index gather
    float ssum = 0.0f;
#pragma unroll
    for (int c = 0; c < NCLS; ++c) ssum += __expf(v[c] - m);
    const float lp = xt - m - __logf(ssum);  // log p(target) <= 0

    logp_out[p] = valid ? lp : 1.0f;         // sentinel > 0 marks ignored
    if (valid) {
      const float prob = __expf(lp);
      int bin = (int)(prob * (float)NB);
      bin = bin < 0 ? 0 : (bin > NB - 1 ? NB - 1 : bin);
      atomicAdd(&hist[bin], 1u);             // ds_add_u32
    }
  }

  __syncthreads();
#pragma unroll
  for (int i = 0; i < NB / BLK; ++i) {
    const unsigned h = hist[tid + i * BLK];
    if (h) atomicAdd(&ghist[tid + i * BLK], h);
  }
}

// ---------------- kernel 0: zero histogram + scalars ----------------
__global__ __launch_bounds__(BLK) void k_init(unsigned* __restrict__ wsu) {
  const int tid = threadIdx.x;
#pragma unroll
  for (int i = 0; i < NB / BLK; ++i) wsu[WS_HIST + tid + i * BLK] = 0u;
  if (tid < 8) wsu[tid] = 0u;
}

// ---------------- kernel 2: quantile -> threshold ----------------
__global__ __launch_bounds__(BLK) void k_thresh(const unsigned* __restrict__ hist,
                                                float* __restrict__ wsf) {
  __shared__ unsigned pref[BLK];
  __shared__ unsigned total;
  __shared__ unsigned pick;
  const int tid = threadIdx.x;
  const int per = NB / BLK;                  // 8 bins per thread
  unsigned s = 0;
  for (int i = 0; i < per; ++i) s += hist[tid * per + i];
  pref[tid] = s;
  __syncthreads();
  if (tid == 0) {
    unsigned run = 0;
    for (int i = 0; i < BLK; ++i) { unsigned v = pref[i]; pref[i] = run; run += v; }
    total = run;
    pick = NB - 1;
  }
  __syncthreads();
  const unsigned nv = total;
  long k = (long)MIN_KEPT;
  if ((long)nv - 1 < k) k = (long)nv - 1;
  if (k < 0) k = 0;
  const unsigned need = (unsigned)k + 1u;    // first bin with cumsum >= need
  unsigned run = pref[tid];
  for (int i = 0; i < per; ++i) {
    run += hist[tid * per + i];
    if (run >= need) { atomicMin(&pick, (unsigned)(tid * per + i)); break; }
  }
  __syncthreads();
  if (tid == 0) {
    const float minv = ((float)pick + 0.5f) * (1.0f / (float)NB);
    wsf[WS_THR] = fmaxf(minv, THRESH_F);
    ((unsigned*)wsf)[WS_NVALID] = nv;
  }
}

// ---------------- kernel 3: selection + fixed-slot partial sums ----------------
__global__ __launch_bounds__(BLK) void k_select(const float* __restrict__ logp,
                                                const float* __restrict__ wsf,
                                                float* __restrict__ psum,
                                                unsigned* __restrict__ pcnt) {
  const float thr = wsf[WS_THR];
  float accs = 0.0f;
  unsigned accc = 0u;
  const int nthr = GRID3 * BLK;
  const float4* lp4 = (const float4*)logp;
  const int n4 = NPIX / 4;
  for (int i = blockIdx.x * BLK + threadIdx.x; i < n4; i += nthr) {
    const float4 v = lp4[i];
    const float a[4] = {v.x, v.y, v.z, v.w};
#pragma unroll
    for (int j = 0; j < 4; ++j) {
      const float lv = a[j];
      const bool valid = (lv <= 0.0f);
      const float prob = __expf(lv);
      const bool sel = valid && (prob < thr);
      accs += sel ? -lv : 0.0f;
      accc += sel ? 1u : 0u;
    }
  }
  __shared__ float sred[BLK];
  __shared__ unsigned cred[BLK];
  sred[threadIdx.x] = accs;
  cred[threadIdx.x] = accc;
  __syncthreads();
  for (int off = BLK / 2; off > 0; off >>= 1) {
    if ((int)threadIdx.x < off) {
      sred[threadIdx.x] += sred[threadIdx.x + off];
      cred[threadIdx.x] += cred[threadIdx.x + off];
    }
    __syncthreads();
  }
  if (threadIdx.x == 0) { psum[blockIdx.x] = sred[0]; pcnt[blockIdx.x] = cred[0]; }
}

// ---------------- kernel 4: deterministic final reduction ----------------
__global__ __launch_bounds__(BLK) void k_final(const float* __restrict__ wsf,
                                               const float* __restrict__ psum,
                                               const unsigned* __restrict__ pcnt,
                                               float* __restrict__ out) {
  __shared__ float sred[BLK];
  __shared__ unsigned cred[BLK];
  const int tid = threadIdx.x;
  float s = 0.0f;
  unsigned c = 0u;
  for (int i = tid; i < GRID3; i += BLK) { s += psum[i]; c += pcnt[i]; }
  sred[tid] = s;
  cred[tid] = c;
  __syncthreads();
  for (int off = BLK / 2; off > 0; off >>= 1) {
    if (tid < off) { sred[tid] += sred[tid + off]; cred[tid] += cred[tid + off]; }
    __syncthreads();
  }
  if (tid == 0) {
    const unsigned nv = ((const unsigned*)wsf)[WS_NVALID];
    const unsigned cnt = cred[0];
    const float mean = sred[0] / (float)(cnt ? cnt : 1u);
    out[0] = (nv > 0u) ? mean : 0.0f;        // LOSS_WEIGHT = 1.0
  }
}

// ---------------- host entry ----------------
extern "C" void kernel_launch(void* const* d_in, const int* in_sizes, int n_in,
                              void* d_out, int out_size, void* d_ws, size_t ws_size,
                              hipStream_t stream) {
  const float* score = (const float*)d_in[0];   // [8,19,512,1024] fp32
  const int* target  = (const int*)d_in[1];     // [8,512,1024] int32
  float* out = (float*)d_out;                   // scalar fp32
  float* wsf = (float*)d_ws;
  unsigned* wsu = (unsigned*)d_ws;

  const size_t need = (size_t)(WS_LOGP + NPIX) * 4u;  // ~16.8 MB
  if (ws_size < need) return;                          // scratch too small

  k_init<<<1, BLK, 0, stream>>>(wsu);
  k_ce<<<GRID1, BLK, 0, stream>>>(score, target, wsf + WS_LOGP, wsu + WS_HIST);
  k_thresh<<<1, BLK, 0, stream>>>(wsu + WS_HIST, wsf);
  k_select<<<GRID3, BLK, 0, stream>>>(wsf + WS_LOGP, wsf, wsf + WS_PSUM, wsu + WS_PCNT);
  k_final<<<1, BLK, 0, stream>>>(wsf, wsf + WS_PSUM, wsu + WS_PCNT, out);
}